// StateUpdate_53549652246919
// MI455X (gfx1250) — compile-verified
//
#include <hip/hip_runtime.h>

typedef __attribute__((ext_vector_type(2))) float v2f;
typedef __attribute__((ext_vector_type(8))) float v8f;

#define NG    8192
#define FEAT  128
#define VIN   384   // 128 bonds + 128 sites + 128 state
#define H1W   256
#define H2W   256
#define OUTW  128
#define VSTR  388   // padded LDS stride for v (384 + 4)
#define ASTR  260   // padded LDS stride for activations (256 + 4)

// ---------------------------------------------------------------------------
// Kernel 1: zero the scatter accumulators in workspace
// ---------------------------------------------------------------------------
__global__ __launch_bounds__(256) void zero_ws_kernel(float* __restrict__ ws, int n) {
    int i = blockIdx.x * 256 + threadIdx.x;
    if (i < n) ws[i] = 0.0f;
}

// ---------------------------------------------------------------------------
// Kernel 2: scatter-add rows into per-graph sums + counts.
// One wave32 per row: each lane loads float4 (global_load_b128) and issues
// 4 hardware f32 atomics (global_atomic_add_f32) into the L2-resident 4MB
// accumulator. Lane 0 counts the row.
// ---------------------------------------------------------------------------
__global__ __launch_bounds__(256) void scatter_kernel(const float* __restrict__ rows,
                                                      const int*   __restrict__ idx,
                                                      float* __restrict__ sum,
                                                      float* __restrict__ cnt,
                                                      int n_rows) {
    int row  = blockIdx.x * 8 + (threadIdx.x >> 5);   // 8 waves per block
    int lane = threadIdx.x & 31;
    if (row >= n_rows) return;
    int g = idx[row];
    const float4 v = *(const float4*)(rows + (size_t)row * FEAT + lane * 4);
    float* dst = sum + (size_t)g * FEAT + lane * 4;
    unsafeAtomicAdd(dst + 0, v.x);
    unsafeAtomicAdd(dst + 1, v.y);
    unsafeAtomicAdd(dst + 2, v.z);
    unsafeAtomicAdd(dst + 3, v.w);
    if (lane == 0) unsafeAtomicAdd(cnt + g, 1.0f);
}

// ---------------------------------------------------------------------------
// One 16x16 output tile via V_WMMA_F32_16X16X4_F32, K stepped by 4.
// A fragment: lane L holds A[M = L&15][K = k + 2*(L>>4) .. +1]   (2 VGPRs)
// B fragment: lane L holds B[K = k + 2*(L>>4) .. +1][N = L&15]   (2 VGPRs)
//   with B[kk][n] = W[(n0+n)*K + kk]  (row-major [out,in] weights => v @ W.T)
// C/D: acc[i] = D[M = i + 8*(L>>4)][N = L&15]
// ---------------------------------------------------------------------------
__device__ __forceinline__ v8f gemm_tile(const float* __restrict__ a_base, int astr,
                                         const float* __restrict__ wrow_base, int K,
                                         float bias, int lr, int hi) {
    v8f acc;
#pragma unroll
    for (int i = 0; i < 8; ++i) acc[i] = bias;
    const float* arow = a_base + lr * astr + 2 * hi;
    const float* wrow = wrow_base + 2 * hi;
    for (int k = 0; k < K; k += 4) {
        v2f a; a[0] = arow[k]; a[1] = arow[k + 1];
        v2f b; b[0] = wrow[k]; b[1] = wrow[k + 1];
        acc = __builtin_amdgcn_wmma_f32_16x16x4_f32(false, a, false, b,
                                                    (short)0, acc, false, false);
    }
    return acc;
}

// ---------------------------------------------------------------------------
// Kernel 3: fused mean + concat + 3-layer MLP (ReLU), 16 graph rows per block.
// 256 threads = 8 wave32s. Activations live in LDS (~57 KB / WGP of 320 KB).
// ---------------------------------------------------------------------------
__global__ __launch_bounds__(256) void mlp_kernel(const float* __restrict__ bonds_sum,
                                                  const float* __restrict__ bcnt,
                                                  const float* __restrict__ sites_sum,
                                                  const float* __restrict__ scnt,
                                                  const float* __restrict__ states,
                                                  const float* __restrict__ W1, const float* __restrict__ b1,
                                                  const float* __restrict__ W2, const float* __restrict__ b2,
                                                  const float* __restrict__ W3, const float* __restrict__ b3,
                                                  float* __restrict__ out) {
    __shared__ float vbuf[16 * VSTR];
    __shared__ float a1[16 * ASTR];
    __shared__ float a2[16 * ASTR];

    const int tid = threadIdx.x;
    const int m0  = blockIdx.x * 16;

    // Stage v = [mean(bonds) | mean(sites) | state] for 16 rows into LDS.
    for (int e = tid; e < 16 * VIN; e += 256) {
        int r = e / VIN, c = e % VIN;
        int g = m0 + r;
        float val;
        if (c < 128)       val = bonds_sum[(size_t)g * FEAT + c]         / fmaxf(bcnt[g], 1.0f);
        else if (c < 256)  val = sites_sum[(size_t)g * FEAT + (c - 128)] / fmaxf(scnt[g], 1.0f);
        else               val = states[(size_t)g * FEAT + (c - 256)];
        vbuf[r * VSTR + c] = val;
    }
    __syncthreads();

    const int wave = tid >> 5;
    const int lane = tid & 31;
    const int lr   = lane & 15;   // M or N within tile
    const int hi   = lane >> 4;   // half-wave selector

    // Layer 1: [16x384] @ [384x256]^T -> a1 ; 16 N-tiles over 8 waves (2 each)
    for (int t = wave; t < 16; t += 8) {
        const int n0 = t * 16;
        v8f acc = gemm_tile(vbuf, VSTR, W1 + (size_t)(n0 + lr) * VIN, VIN,
                            b1[n0 + lr], lr, hi);
#pragma unroll
        for (int i = 0; i < 8; ++i) {
            int M = i + hi * 8;
            a1[M * ASTR + n0 + lr] = fmaxf(acc[i], 0.0f);
        }
    }
    __syncthreads();

    // Layer 2: [16x256] @ [256x256]^T -> a2
    for (int t = wave; t < 16; t += 8) {
        const int n0 = t * 16;
        v8f acc = gemm_tile(a1, ASTR, W2 + (size_t)(n0 + lr) * H1W, H1W,
                            b2[n0 + lr], lr, hi);
#pragma unroll
        for (int i = 0; i < 8; ++i) {
            int M = i + hi * 8;
            a2[M * ASTR + n0 + lr] = fmaxf(acc[i], 0.0f);
        }
    }
    __syncthreads();

    // Layer 3: [16x256] @ [256x128]^T -> out ; 8 N-tiles, one per wave
    {
        const int n0 = wave * 16;
        v8f acc = gemm_tile(a2, ASTR, W3 + (size_t)(n0 + lr) * H2W, H2W,
                            b3[n0 + lr], lr, hi);
#pragma unroll
        for (int i = 0; i < 8; ++i) {
            int M = i + hi * 8;
            out[(size_t)(m0 + M) * OUTW + n0 + lr] = fmaxf(acc[i], 0.0f);
        }
    }
}

// ---------------------------------------------------------------------------
// Host launcher
// ---------------------------------------------------------------------------
extern "C" void kernel_launch(void* const* d_in, const int* in_sizes, int n_in,
                              void* d_out, int out_size, void* d_ws, size_t ws_size,
                              hipStream_t stream) {
    const float* sites  = (const float*)d_in[0];
    const float* bonds  = (const float*)d_in[1];
    const float* states = (const float*)d_in[2];
    const int*   g2s    = (const int*)  d_in[3];
    const int*   g2b    = (const int*)  d_in[4];
    const float* W1     = (const float*)d_in[5];
    const float* b1     = (const float*)d_in[6];
    const float* W2     = (const float*)d_in[7];
    const float* b2     = (const float*)d_in[8];
    const float* W3     = (const float*)d_in[9];
    const float* b3     = (const float*)d_in[10];
    float*       out    = (float*)d_out;

    const int n_sites = in_sizes[0] / FEAT;
    const int n_bonds = in_sizes[1] / FEAT;

    // Workspace layout (f32): bonds_sum[NG*128] | sites_sum[NG*128] | bcnt[NG] | scnt[NG]
    float* ws        = (float*)d_ws;
    float* bonds_sum = ws;
    float* sites_sum = ws + (size_t)NG * FEAT;
    float* bcnt      = ws + (size_t)2 * NG * FEAT;
    float* scnt      = bcnt + NG;

    const int n_zero = 2 * NG * FEAT + 2 * NG;
    zero_ws_kernel<<<(n_zero + 255) / 256, 256, 0, stream>>>(ws, n_zero);

    scatter_kernel<<<(n_bonds + 7) / 8, 256, 0, stream>>>(bonds, g2b, bonds_sum, bcnt, n_bonds);
    scatter_kernel<<<(n_sites + 7) / 8, 256, 0, stream>>>(sites, g2s, sites_sum, scnt, n_sites);

    mlp_kernel<<<NG / 16, 256, 0, stream>>>(bonds_sum, bcnt, sites_sum, scnt, states,
                                            W1, b1, W2, b2, W3, b3, out);
}